// GCNEncoder_42442866819845
// MI455X (gfx1250) — compile-verified
//
#include <hip/hip_runtime.h>

typedef __attribute__((ext_vector_type(2))) float v2f;
typedef __attribute__((ext_vector_type(8))) float v8f;

#define FIN  16
#define FOUT 128

// ---------------- degree / normalization ----------------
__global__ void k_init_deg(float* __restrict__ deg, int n) {
  int i = blockIdx.x * blockDim.x + threadIdx.x;
  if (i < n) deg[i] = 1.0f;  // self-loop contributes 1 to every node's degree
}

__global__ void k_count_deg(const int* __restrict__ col, float* __restrict__ deg, int nE) {
  int e = blockIdx.x * blockDim.x + threadIdx.x;
  if (e < nE) atomicAdd(&deg[col[e]], 1.0f);
}

__global__ void k_rsqrt_deg(float* __restrict__ deg, int n) {
  int i = blockIdx.x * blockDim.x + threadIdx.x;
  if (i < n) deg[i] = rsqrtf(deg[i]);  // deg >= 1 always (self-loop)
}

// ---------------- GEMM1: H = X[N,16] @ W[16,128] (fp32 WMMA) ----------------
// block = 256 threads = 8 waves; each block does a 16-row M-tile,
// each wave one 16-col N-tile. K = 16 -> 4 wmma_f32_16x16x4_f32 steps.
__global__ __launch_bounds__(256) void k_gemm1(const float* __restrict__ X,
                                               const float* __restrict__ W,
                                               float* __restrict__ H) {
  const int lane = threadIdx.x & 31;
  const int wv   = threadIdx.x >> 5;   // N-tile 0..7
  const int half = lane >> 4;          // 0 / 1
  const int l15  = lane & 15;
  const int m0   = blockIdx.x * 16;
  const int col  = wv * 16 + l15;

  const float* xr = X + (size_t)(m0 + l15) * FIN;
  v8f acc = {};
#pragma unroll
  for (int kk = 0; kk < 4; ++kk) {
    const int ka = kk * 4 + half * 2;  // K index for (vgpr0, lane-half)
    v2f a, b;
    a.x = xr[ka];
    a.y = xr[ka + 1];
    b.x = W[(size_t)ka * FOUT + col];
    b.y = W[(size_t)(ka + 1) * FOUT + col];
    acc = __builtin_amdgcn_wmma_f32_16x16x4_f32(false, a, false, b,
                                                (short)0, acc, false, false);
  }
  float* out = H + (size_t)(m0 + half * 8) * FOUT + col;
#pragma unroll
  for (int g = 0; g < 8; ++g) out[(size_t)g * FOUT] = acc[g];
}

// ---------------- GEMM2: H = A[N,128] @ W[128,128] (fp32 WMMA, LDS-staged A) ----
#define LDS_STRIDE 132   // pad 128 -> 132 floats: column reads hit distinct banks
__global__ __launch_bounds__(256) void k_gemm2(const float* __restrict__ A,
                                               const float* __restrict__ W,
                                               float* __restrict__ H) {
  __shared__ float lds[16 * LDS_STRIDE];
  const int m0 = blockIdx.x * 16;

  // stage the 16x128 A tile once per block (float4 loads)
  for (int t = threadIdx.x; t < 16 * 32; t += 256) {
    const int r  = t >> 5;
    const int c4 = (t & 31) * 4;
    const float4 v = *(const float4*)(A + (size_t)(m0 + r) * FOUT + c4);
    float* dst = &lds[r * LDS_STRIDE + c4];
    dst[0] = v.x; dst[1] = v.y; dst[2] = v.z; dst[3] = v.w;
  }
  __syncthreads();

  const int lane = threadIdx.x & 31;
  const int wv   = threadIdx.x >> 5;
  const int half = lane >> 4;
  const int l15  = lane & 15;
  const int col  = wv * 16 + l15;

  v8f acc = {};
#pragma unroll 4
  for (int kk = 0; kk < 32; ++kk) {
    const int ka = kk * 4 + half * 2;
    v2f a, b;
    a.x = lds[l15 * LDS_STRIDE + ka];
    a.y = lds[l15 * LDS_STRIDE + ka + 1];
    b.x = W[(size_t)ka * FOUT + col];
    b.y = W[(size_t)(ka + 1) * FOUT + col];
    acc = __builtin_amdgcn_wmma_f32_16x16x4_f32(false, a, false, b,
                                                (short)0, acc, false, false);
  }
  float* out = H + (size_t)(m0 + half * 8) * FOUT + col;
#pragma unroll
  for (int g = 0; g < 8; ++g) out[(size_t)g * FOUT] = acc[g];
}

// ---------------- aggregation ----------------
// OUT[i,:] = H[i,:] * dis[i]^2   (self-loop term; also fully initializes OUT)
__global__ void k_selfloop(const float* __restrict__ dis, const float* __restrict__ H,
                           float* __restrict__ OUT, int n) {
  int idx = blockIdx.x * blockDim.x + threadIdx.x;   // thread per (node, 4 feats)
  int i  = idx >> 5;
  int f4 = (idx & 31) * 4;
  if (i >= n) return;
  float s = dis[i]; s *= s;
  const float4 hv = *(const float4*)(H + (size_t)i * FOUT + f4);
  float4 o = make_float4(hv.x * s, hv.y * s, hv.z * s, hv.w * s);
  *(float4*)(OUT + (size_t)i * FOUT + f4) = o;
}

// per edge e: OUT[row[e],:] += H[col[e],:] * dis[row]*dis[col]
__global__ void k_edge_agg(const int* __restrict__ row, const int* __restrict__ col,
                           const float* __restrict__ dis, const float* __restrict__ H,
                           float* __restrict__ OUT, int nE) {
  int idx = blockIdx.x * blockDim.x + threadIdx.x;   // 32 threads per edge
  int e  = idx >> 5;
  int f4 = (idx & 31) * 4;
  if (e >= nE) return;
  const int r = row[e];
  const int c = col[e];
  const float nrm = dis[r] * dis[c];
  const float4 hv = *(const float4*)(H + (size_t)c * FOUT + f4);
  float* o = OUT + (size_t)r * FOUT + f4;
  atomicAdd(o + 0, hv.x * nrm);
  atomicAdd(o + 1, hv.y * nrm);
  atomicAdd(o + 2, hv.z * nrm);
  atomicAdd(o + 3, hv.w * nrm);
}

__global__ void k_bias_relu(float* __restrict__ H, const float* __restrict__ b, int total) {
  int idx = blockIdx.x * blockDim.x + threadIdx.x;
  if (idx < total) {
    float v = H[idx] + b[idx & (FOUT - 1)];
    H[idx] = v > 0.0f ? v : 0.0f;
  }
}

__global__ void k_bias(float* __restrict__ H, const float* __restrict__ b, int total) {
  int idx = blockIdx.x * blockDim.x + threadIdx.x;
  if (idx < total) H[idx] += b[idx & (FOUT - 1)];
}

// ---------------- launcher ----------------
extern "C" void kernel_launch(void* const* d_in, const int* in_sizes, int n_in,
                              void* d_out, int out_size, void* d_ws, size_t ws_size,
                              hipStream_t stream) {
  const float* x  = (const float*)d_in[0];
  const float* W1 = (const float*)d_in[1];
  const float* b1 = (const float*)d_in[2];
  const float* W2 = (const float*)d_in[3];
  const float* b2 = (const float*)d_in[4];
  const int*   ei = (const int*)d_in[5];

  const int n  = in_sizes[0] / FIN;   // 100000 (divisible by 16)
  const int nE = in_sizes[5] / 2;     // 640000
  const int* row = ei;        // edge_index[0] = destinations
  const int* col = ei + nE;   // edge_index[1] = sources

  float* dis  = (float*)d_ws;                       // n floats (holds deg, then deg^-1/2)
  float* bufA = dis + (((size_t)n + 255) & ~(size_t)255);
  float* bufB = bufA + (size_t)n * FOUT;
  float* out  = (float*)d_out;

  const dim3 blk(256);
  const int gN   = (n + 255) / 256;
  const int gE   = (nE + 255) / 256;
  const int gN32 = (n * 32 + 255) / 256;
  const int gE32 = (nE * 32 + 255) / 256;
  const int gNF  = (n * FOUT + 255) / 256;
  const int gMM  = n / 16;

  // normalization: dis = rsqrt(1 + count(col))
  k_init_deg <<<gN, blk, 0, stream>>>(dis, n);
  k_count_deg<<<gE, blk, 0, stream>>>(col, dis, nE);
  k_rsqrt_deg<<<gN, blk, 0, stream>>>(dis, n);

  // ---- layer 1 ----
  k_gemm1    <<<gMM,  blk, 0, stream>>>(x, W1, bufA);            // bufA = x@W1
  k_selfloop <<<gN32, blk, 0, stream>>>(dis, bufA, bufB, n);     // bufB = selfloop init
  k_edge_agg <<<gE32, blk, 0, stream>>>(row, col, dis, bufA, bufB, nE);
  k_bias_relu<<<gNF,  blk, 0, stream>>>(bufB, b1, n * FOUT);     // bufB = relu(agg + b1)

  // ---- layer 2 ---- (reuse bufA for h2)
  k_gemm2    <<<gMM,  blk, 0, stream>>>(bufB, W2, bufA);         // bufA = h@W2
  k_selfloop <<<gN32, blk, 0, stream>>>(dis, bufA, out, n);      // out fully initialized
  k_edge_agg <<<gE32, blk, 0, stream>>>(row, col, dis, bufA, out, nE);
  k_bias     <<<gNF,  blk, 0, stream>>>(out, b2, n * FOUT);
}